// Hadamard_2448131359091
// MI455X (gfx1250) — compile-verified
//
#include <hip/hip_runtime.h>

// Batched FWHT (N=2^14, BATCH=4096) recast as Y = (1/128) * H128 * X * H128
// per batch row (X = row viewed as 128x128 f32, H128[r][c] = (-1)^popcount(r&c)).
// Both GEMMs run on V_WMMA_F32_16X16X4_F32 (exact f32 datapath; H synthesized
// per lane via popcount -> sign bit, never loaded from memory).
// Register-blocked: each wave holds 8 accumulator tiles (its full 16x128 output
// band), so the A operand (H signs in phase 1, T loads in phase 2) is produced
// once per k-step and amortized over 8 WMMAs.
// X staging uses GLOBAL_LOAD_ASYNC_TO_LDS_B128 (ASYNCcnt path) when available.
// HBM traffic: 256 MB in + 256 MB out -> ~22 us floor at 23.3 TB/s.

typedef __attribute__((ext_vector_type(2))) float v2f;
typedef __attribute__((ext_vector_type(8))) float v8f;
typedef __attribute__((ext_vector_type(4))) int   v4i;

#define LDS_STRIDE 132   // 128 + 4 floats pad -> conflict-friendly row stride

#if defined(__has_builtin)
#if __has_builtin(__builtin_amdgcn_global_load_async_to_lds_b128) && \
    __has_builtin(__builtin_amdgcn_s_wait_asynccnt)
#define HAVE_ASYNC_LDS 1
#endif
#endif

__device__ __forceinline__ float hsign(int r, int c) {
    // +/-1.0f from parity of popcount(r & c): OR parity into the sign bit.
    unsigned p = (unsigned)(__builtin_popcount(r & c) & 1);
    union { unsigned u; float f; } x;
    x.u = 0x3F800000u | (p << 31);
    return x.f;
}

__global__ __launch_bounds__(256)
void fwht_wmma_kernel(const float* __restrict__ phi, float* __restrict__ out) {
    extern __shared__ float lds[];
    float* Xs = lds;                        // [128][LDS_STRIDE]
    float* Ts = lds + 128 * LDS_STRIDE;     // [128][LDS_STRIDE]

    const int t    = threadIdx.x;
    const int lane = t & 31;
    const int wave = t >> 5;                // 0..7 (wave32)
    const long base = (long)blockIdx.x * 16384;

    // ---- Stage X (128x128 of this batch row) into LDS, 16B chunks ----
    #pragma unroll
    for (int i = 0; i < 16; ++i) {
        int c4   = i * 256 + t;             // 0..4095 float4 chunks
        int row  = c4 >> 5;                 // 32 float4 per 128-float row
        int col4 = c4 & 31;
        const float* gp = phi + base + (long)c4 * 4;
        float* lp = Xs + row * LDS_STRIDE + col4 * 4;
#ifdef HAVE_ASYNC_LDS
        __builtin_amdgcn_global_load_async_to_lds_b128(
            (__attribute__((address_space(1))) v4i*)gp,
            (__attribute__((address_space(3))) v4i*)lp, 0, 0);
#else
        *(float4*)lp = *(const float4*)gp;
#endif
    }
#ifdef HAVE_ASYNC_LDS
    __builtin_amdgcn_s_wait_asynccnt(0);
#endif
    __syncthreads();

    // WMMA f32 16x16x4 register layouts (wave32):
    //   A: lanes 0-15 = M rows, v0/v1 = K{0,1}; lanes 16-31 same M, K{2,3}
    //   B: lanes 0-15 = N cols, v0/v1 = K{0,1}; lanes 16-31 same N, K{2,3}
    //   C/D: vj -> (M=j, N=lane) lanes 0-15, (M=j+8, N=lane-16) lanes 16-31
    const int lm   = lane & 15;             // M (A) / N (B,C,D) within tile
    const int lk   = (lane >> 4) * 2;       // K sub-offset: 0 or 2
    const int mhi  = (lane >> 4) << 3;      // +8 M offset for upper half-wave C/D
    const int row0 = wave * 16;             // this wave's output row band

    // ---- Phase 1: T = H * X  (A = H once per k-step, B = X from LDS) ----
    {
        v8f acc[8] = {};
        for (int k = 0; k < 128; k += 4) {
            v2f a;
            a.x = hsign(row0 + lm, k + lk);
            a.y = hsign(row0 + lm, k + lk + 1);
            const float* xr = Xs + (k + lk) * LDS_STRIDE + lm;
            #pragma unroll
            for (int cb = 0; cb < 8; ++cb) {
                v2f bm;
                bm.x = xr[cb * 16];                 // row k+lk
                bm.y = xr[cb * 16 + LDS_STRIDE];    // row k+lk+1 -> ds_load_2addr
                acc[cb] = __builtin_amdgcn_wmma_f32_16x16x4_f32(
                    false, a, false, bm, (short)0, acc[cb], false, false);
            }
        }
        #pragma unroll
        for (int cb = 0; cb < 8; ++cb) {
            #pragma unroll
            for (int j = 0; j < 8; ++j) {
                int r = row0 + j + mhi;
                Ts[r * LDS_STRIDE + cb * 16 + lm] = acc[cb][j];
            }
        }
    }
    __syncthreads();

    // ---- Phase 2: Y = (1/128) * T * H  (A = T once per k-step, B = H signs) ----
    {
        v8f acc[8] = {};
        for (int k = 0; k < 128; k += 4) {
            v2f a;
            const float* tp = Ts + (row0 + lm) * LDS_STRIDE + k + lk;
            a.x = tp[0];                    // contiguous pair -> ds_load_b64
            a.y = tp[1];
            #pragma unroll
            for (int cb = 0; cb < 8; ++cb) {
                v2f bm;
                bm.x = hsign(k + lk,     cb * 16 + lm);
                bm.y = hsign(k + lk + 1, cb * 16 + lm);
                acc[cb] = __builtin_amdgcn_wmma_f32_16x16x4_f32(
                    false, a, false, bm, (short)0, acc[cb], false, false);
            }
        }
        #pragma unroll
        for (int cb = 0; cb < 8; ++cb) {
            #pragma unroll
            for (int j = 0; j < 8; ++j) {
                int r = row0 + j + mhi;
                out[base + (long)r * 128 + cb * 16 + lm] = acc[cb][j] * 0.0078125f;
            }
        }
    }
}

extern "C" void kernel_launch(void* const* d_in, const int* in_sizes, int n_in,
                              void* d_out, int out_size, void* d_ws, size_t ws_size,
                              hipStream_t stream) {
    (void)n_in; (void)d_ws; (void)ws_size; (void)out_size;
    const float* phi = (const float*)d_in[0];
    float* out = (float*)d_out;

    const int N = 16384;                         // 2^14 per row
    const int batch = in_sizes[0] / N;           // 4096
    const size_t shmem = (size_t)2 * 128 * LDS_STRIDE * sizeof(float); // 135168 B

    // Allow >64KB dynamic LDS (WGP has 320KB on gfx1250).
    (void)hipFuncSetAttribute((const void*)fwht_wmma_kernel,
                              hipFuncAttributeMaxDynamicSharedMemorySize,
                              (int)shmem);

    fwht_wmma_kernel<<<dim3(batch), dim3(256), shmem, stream>>>(phi, out);
}